// Hbv_1_1p_23596550324554
// MI455X (gfx1250) — compile-verified
//
#include <hip/hip_runtime.h>
#include <stdint.h>

#define NSTEPS 730
#define NGRID  8000
#define UHL    15
#define BLOCK  160                 // 5 wave32s
#define NBLK   (NGRID / BLOCK)     // 50 blocks, exact
#define DEPTH  8                   // TDM pipeline depth (ring buffers)
#define NEARZERO 1e-5f

typedef __attribute__((ext_vector_type(4))) unsigned v4u;
typedef __attribute__((ext_vector_type(8))) unsigned v8u;

// ---- Tensor Data Mover: 1D row (nelem f32) global -> LDS, D# per CDNA5 ISA ----
__device__ __forceinline__ void tdm_load_row(uint32_t lds_byte_addr, uint64_t gaddr,
                                             uint32_t nelem) {
  v4u g0; v8u g1;
  g0[0] = 1u;                                        // count=1, user descriptor
  g0[1] = lds_byte_addr;                             // lds_addr [63:32]
  g0[2] = (uint32_t)gaddr;                           // global_addr lo  [95:64]
  g0[3] = (uint32_t)(gaddr >> 32) | (2u << 30);      // addr hi + type=2 [127:126]
  g1[0] = 2u << 16;                                  // data_size=2 (4B), no mask/pad
  g1[1] = (nelem & 0xFFFFu) << 16;                   // tensor_dim0 lo16 @ [31:16]
  g1[2] = (nelem >> 16) | (1u << 16);                // dim0 hi16 ; tensor_dim1=1
  g1[3] = (nelem & 0xFFFFu) << 16;                   // tile_dim0 @ [127:112]
  g1[4] = 1u;                                        // tile_dim1=1, tile_dim2=0
  g1[5] = nelem;                                     // tensor_dim0_stride lo32
  g1[6] = 0u;                                        // stride hi / dim1_stride lo
  g1[7] = 0u;
  asm volatile("tensor_load_to_lds %0, %1" :: "s"(g0), "s"(g1) : "memory");
}

__device__ __forceinline__ float fpowf(float x, float b) {  // x > 0 guaranteed
  return __expf(b * __logf(x));
}
__device__ __forceinline__ float sigmoidf(float x) {
  return 1.0f / (1.0f + __expf(-x));
}

__global__ __launch_bounds__(BLOCK, 1)
void hbv_1_1p_fused(const float* __restrict__ x_phy,
                    const float* __restrict__ params,
                    float4* __restrict__ out) {
  __shared__ float smem[DEPTH][BLOCK * 3];           // 8 x 1.92 KB ring
  const int      g    = blockIdx.x * BLOCK + threadIdx.x;
  const unsigned wid  = threadIdx.x >> 5;
  const unsigned lane = threadIdx.x & 31;
  const float*   sp   = &smem[0][0];

  // wave-uniform TDM addressing: each wave owns a 32-cell (96-float) slice
  const uint32_t lbase = (uint32_t)(uintptr_t)&smem[0][0];
  const uint32_t slice = wid * 96u;                                   // floats
  const uint64_t gwbase = (uint64_t)(uintptr_t)x_phy +
      (uint64_t)((blockIdx.x * BLOCK + (wid << 5)) * 3u) * 4u;
  const uint64_t trow = (uint64_t)NGRID * 3u * 4u;                    // bytes/step

  // prime the TDM ring with rows 0..DEPTH-1 ASAP (hide latency behind setup)
  #pragma unroll
  for (int i = 0; i < DEPTH; ++i)
    tdm_load_row(lbase + (uint32_t)(i * BLOCK * 3 + (int)slice) * 4u,
                 gwbase + (uint64_t)i * trow, 96u);

  // ---- static parameters: sigmoid(last-step row), descaled to PHY_BOUNDS ----
  const float* pr = params + ((size_t)(NSTEPS - 1) * NGRID + g) * 16;
  float raw[16];
  #pragma unroll
  for (int i = 0; i < 16; ++i) raw[i] = sigmoidf(pr[i]);
  const float parBETA   = 1.0f   + raw[0]  * 5.0f;
  const float parFC     = 50.0f  + raw[1]  * 950.0f;
  const float parK0     = 0.05f  + raw[2]  * 0.85f;
  const float parK1     = 0.01f  + raw[3]  * 0.49f;
  const float parK2     = 0.001f + raw[4]  * 0.199f;
  const float parLP     = 0.2f   + raw[5]  * 0.8f;
  const float parPERC   =          raw[6]  * 10.0f;
  const float parUZL    =          raw[7]  * 100.0f;
  const float parTT     = -2.5f  + raw[8]  * 5.0f;
  const float parCFMAX  = 0.5f   + raw[9]  * 9.5f;
  const float parCFR    =          raw[10] * 0.1f;
  const float parCWH    =          raw[11] * 0.2f;
  const float parBETAET = 0.3f   + raw[12] * 4.7f;
  const float parC      =          raw[13];
  const float invFC   = 1.0f / parFC;
  const float invLPFC = 1.0f / (parLP * parFC);
  const float CFRCF   = parCFR * parCFMAX;

  // ---- gamma unit hydrograph weights (log-space, lgamma once) ----
  float w[UHL];
  {
    const float aa = fmaxf(raw[14] * 2.9f, 0.0f) + 0.1f;
    const float th = fmaxf(raw[15] * 6.5f, 0.0f) + 0.5f;
    const float ldenom = lgammaf(aa) + aa * __logf(th);
    float s = 0.0f;
    #pragma unroll
    for (int k = 0; k < UHL; ++k) {
      const float tk = (float)k + 0.5f;
      w[k] = __expf((aa - 1.0f) * __logf(tk) - tk / th - ldenom);
      s += w[k];
    }
    const float inv = 1.0f / s;
    #pragma unroll
    for (int k = 0; k < UHL; ++k) w[k] *= inv;
  }

  // ---- state + routing ring accumulators (all registers) ----
  float SNOWPACK = 1e-3f, MELTWATER = 1e-3f, SM = 1e-3f, SUZ = 1e-3f, SLZ = 1e-3f;
  float acc[4][UHL];
  #pragma unroll
  for (int c = 0; c < 4; ++c)
    #pragma unroll
    for (int i = 0; i < UHL; ++i) acc[c][i] = 0.0f;

  for (int t = 0; t < NSTEPS; ++t) {
    // TDM completes in-order per wave: <=DEPTH-1 outstanding  =>  row t landed
    if (t <= NSTEPS - DEPTH) __builtin_amdgcn_s_wait_tensorcnt(DEPTH - 1);
    else                     __builtin_amdgcn_s_wait_tensorcnt(0);
    const int buf  = t & (DEPTH - 1);
    const int base = buf * BLOCK * 3 + (int)slice + (int)lane * 3;
    const float Pt   = sp[base + 0];
    const float Tt   = sp[base + 1];
    const float PETt = sp[base + 2];

    // forcings now in VGPRs -> immediately recycle this buffer for row t+DEPTH
    asm volatile("s_wait_dscnt 0" ::: "memory");
    if (t + DEPTH < NSTEPS)
      tdm_load_row(lbase + (uint32_t)(buf * BLOCK * 3 + (int)slice) * 4u,
                   gwbase + (uint64_t)(t + DEPTH) * trow, 96u);

    // ---- HBV-1.1p step ----
    const float RAIN = (Tt >= parTT) ? Pt : 0.0f;
    const float SNOW = (Tt <  parTT) ? Pt : 0.0f;
    SNOWPACK += SNOW;
    const float melt = fminf(fmaxf(parCFMAX * (Tt - parTT), 0.0f), SNOWPACK);
    MELTWATER += melt;  SNOWPACK -= melt;
    const float refr = fminf(fmaxf(CFRCF * (parTT - Tt), 0.0f), MELTWATER);
    SNOWPACK += refr;   MELTWATER -= refr;
    const float tosoil = fmaxf(MELTWATER - parCWH * SNOWPACK, 0.0f);
    MELTWATER -= tosoil;
    const float soil_wet = fminf(fpowf(SM * invFC, parBETA), 1.0f);
    const float recharge = (RAIN + tosoil) * soil_wet;
    SM += RAIN + tosoil - recharge;
    const float excess = fmaxf(SM - parFC, 0.0f);
    SM -= excess;
    const float capil = fminf(SLZ, parC * SLZ * (1.0f - fminf(SM * invFC, 1.0f)));
    SM  = fmaxf(SM + capil, NEARZERO);
    SLZ = fmaxf(SLZ - capil, NEARZERO);
    const float evapf = fpowf(fminf(fmaxf(SM * invLPFC, 0.0f), 1.0f), parBETAET);
    const float ET = fminf(PETt * evapf, SM);
    SM = fmaxf(SM - ET, NEARZERO);
    SUZ += recharge + excess;
    const float perc = fminf(SUZ, parPERC);
    SUZ -= perc;
    const float Q0 = parK0 * fmaxf(SUZ - parUZL, 0.0f);
    SUZ -= Q0;
    const float Q1 = parK1 * SUZ;
    SUZ -= Q1;
    SLZ += perc;
    const float Q2 = parK2 * SLZ;
    SLZ -= Q2;
    const float Qs = Q0 + Q1 + Q2;

    // ---- fused routing: shift-and-accumulate ring (60 FMAs, no moves) ----
    const float q[4] = {Qs, Q0, Q1, Q2};
    float4 o;
    o.x = fmaf(w[0], q[0], acc[0][0]);
    o.y = fmaf(w[0], q[1], acc[1][0]);
    o.z = fmaf(w[0], q[2], acc[2][0]);
    o.w = fmaf(w[0], q[3], acc[3][0]);
    #pragma unroll
    for (int c = 0; c < 4; ++c) {
      #pragma unroll
      for (int i = 0; i < UHL - 1; ++i)
        acc[c][i] = fmaf(w[i + 1], q[c], acc[c][i + 1]);
      acc[c][UHL - 1] = 0.0f;
    }

    out[(size_t)t * NGRID + g] = o;   // global_store_b128, coalesced
  }
}

extern "C" void kernel_launch(void* const* d_in, const int* in_sizes, int n_in,
                              void* d_out, int out_size, void* d_ws, size_t ws_size,
                              hipStream_t stream) {
  const float* x_phy  = (const float*)d_in[0];   // [730, 8000, 3] f32
  const float* params = (const float*)d_in[1];   // [730, 8000, 16] f32
  float4* out = (float4*)d_out;                  // [730, 8000, 4] f32
  hbv_1_1p_fused<<<dim3(NBLK), dim3(BLOCK), 0, stream>>>(x_phy, params, out);
  (void)in_sizes; (void)n_in; (void)out_size; (void)d_ws; (void)ws_size;
}